// BilinearInteractionLayer_5669356835977
// MI455X (gfx1250) — compile-verified
//
#include <hip/hip_runtime.h>
#include <stdint.h>

#define F_FEAT  30
#define D_DIM   128
#define N_PAIRS 435
#define M_TILE  64      // batch rows per workgroup
#define K_CHUNK 32      // K depth per LDS stage
#define NCHUNK  (D_DIM / K_CHUNK)   // 4
#define TSTR    36      // word stride (32 data + 4 pad): conflict-free b64 reads, 16B-aligned rows
#define CSTR    132     // word stride for C write-back tile (conflict-free scatter, 16B-aligned rows)

#define WT_BYTES (D_DIM * TSTR * 4)     // 18,432 B per buffer
#define A_BYTES  (M_TILE * TSTR * 4)    //  9,216 B per buffer
#define SMEM_BYTES (2 * WT_BYTES + 2 * A_BYTES)   // 55,296 B (C tile 33,792 B aliases this)

typedef __attribute__((ext_vector_type(2))) float v2f;
typedef __attribute__((ext_vector_type(4))) float v4f;
typedef __attribute__((ext_vector_type(8))) float v8f;

__global__ __launch_bounds__(256)
void bilinear_wmma_kernel(const float* __restrict__ emb,
                          const float* __restrict__ W,
                          float* __restrict__ out)
{
    __shared__ __attribute__((aligned(16))) char smem[SMEM_BYTES];
    float* sWT0 = (float*)smem;                       // 2 x W^T[e][k] chunk buffers
    float* sA0  = (float*)(smem + 2 * WT_BYTES);      // 2 x A[m][k]   chunk buffers

    const int p  = blockIdx.y;
    const int b0 = blockIdx.x * M_TILE;

    // decode pair index p -> (fi, fj) for combinations(range(30), 2)
    int fi = 0, rem = p, cnt = F_FEAT - 1;
    while (rem >= cnt) { rem -= cnt; ++fi; --cnt; }
    const int fj = fi + 1 + rem;

    const int tid  = threadIdx.x;
    const int lane = tid & 31;
    const int wv   = tid >> 5;        // wave 0..7
    const int mt   = wv & 3;          // M tile 0..3
    const int ng   = wv >> 2;         // N group: tiles ng, ng+2, ng+4, ng+6
    const int l15  = lane & 15;
    const int kb   = (lane >> 4) * 2; // A/B K sub-offset per half-wave
    const int mrow = mt * 16 + l15;
    const int nc0  = ng * 16 + l15;   // columns nc0 + 32*t, t = 0..3

    const float* Wp = W + (size_t)p * (D_DIM * D_DIM);

    // ---- async DMA staging of one K-chunk into buffer `buf` ----
    auto stage = [&](int kc, int buf) {
        // W rows [kc*32, +32) transposed into sWT[e*TSTR + k]:
        // lanes read 128B contiguous global per instruction (fixed k, e = e0 + lane)
        const float* Wc = Wp + (size_t)kc * K_CHUNK * D_DIM;
        float* dWT = sWT0 + buf * (D_DIM * TSTR);
        #pragma unroll
        for (int i = 0; i < 16; ++i) {
            int s = wv * 16 + i;            // 0..127
            int k = s & 31;
            int e = ((s >> 5) << 5) + lane;
            uint64_t g = (uint64_t)(uintptr_t)(Wc + (size_t)k * D_DIM + e);
            uint32_t l = (uint32_t)(uintptr_t)&dWT[e * TSTR + k];
            asm volatile("global_load_async_to_lds_b32 %0, %1, off"
                         :: "v"(l), "v"(g) : "memory");
        }
        // A: 64 rows x 32 K-cols of v_i, row-major, b128 per lane
        float* dA = sA0 + buf * (M_TILE * TSTR);
        #pragma unroll
        for (int t = 0; t < 2; ++t) {
            int idx = wv * 64 + t * 32 + lane;  // float4 index 0..511
            int r   = idx >> 3;
            int c4  = idx & 7;
            const float* src = emb + ((size_t)(b0 + r) * F_FEAT + fi) * D_DIM
                                   + kc * K_CHUNK + c4 * 4;
            uint64_t g = (uint64_t)(uintptr_t)src;
            uint32_t l = (uint32_t)(uintptr_t)&dA[r * TSTR + c4 * 4];
            asm volatile("global_load_async_to_lds_b128 %0, %1, off"
                         :: "v"(l), "v"(g) : "memory");
        }
    };

    v8f acc0 = {}, acc1 = {}, acc2 = {}, acc3 = {};

    stage(0, 0);
    asm volatile("s_wait_asynccnt 0x0" ::: "memory");
    __syncthreads();

    for (int kc = 0; kc < NCHUNK; ++kc) {
        const int buf = kc & 1;
        if (kc + 1 < NCHUNK) stage(kc + 1, buf ^ 1);   // DMA overlapped with WMMA

        const float* bWT = sWT0 + buf * (D_DIM * TSTR);
        const float* bA  = sA0  + buf * (M_TILE * TSTR);
        #pragma unroll
        for (int kk = 0; kk < K_CHUNK / 4; ++kk) {
            int krow = kk * 4 + kb;
            v2f a   = *(const v2f*)&bA [mrow * TSTR + krow];
            v2f b0v = *(const v2f*)&bWT[(nc0     ) * TSTR + krow];
            v2f b1v = *(const v2f*)&bWT[(nc0 + 32) * TSTR + krow];
            v2f b2v = *(const v2f*)&bWT[(nc0 + 64) * TSTR + krow];
            v2f b3v = *(const v2f*)&bWT[(nc0 + 96) * TSTR + krow];
            acc0 = __builtin_amdgcn_wmma_f32_16x16x4_f32(false, a, false, b0v, (short)0, acc0, false, false);
            acc1 = __builtin_amdgcn_wmma_f32_16x16x4_f32(false, a, false, b1v, (short)0, acc1, false, false);
            acc2 = __builtin_amdgcn_wmma_f32_16x16x4_f32(false, a, false, b2v, (short)0, acc2, false, false);
            acc3 = __builtin_amdgcn_wmma_f32_16x16x4_f32(false, a, false, b3v, (short)0, acc3, false, false);
        }

        if (kc + 1 < NCHUNK) {
            asm volatile("s_wait_asynccnt 0x0" ::: "memory");
            __syncthreads();
        }
    }

    // ---- epilogue: transpose C through LDS, multiply by v_j, stream out NT ----
    __syncthreads();                      // compute buffers dead; alias C tile over them
    float* sC = (float*)smem;             // [64][CSTR]

    const int halo = (lane >> 4) * 8;     // C/D layout: upper half-wave holds M+8
    #pragma unroll
    for (int r = 0; r < 8; ++r) {
        int m = mt * 16 + r + halo;
        sC[m * CSTR + nc0     ] = acc0[r];
        sC[m * CSTR + nc0 + 32] = acc1[r];
        sC[m * CSTR + nc0 + 64] = acc2[r];
        sC[m * CSTR + nc0 + 96] = acc3[r];
    }
    __syncthreads();

    // each wave streams whole rows: 512B contiguous per store instruction
    #pragma unroll
    for (int it = 0; it < 8; ++it) {
        int idx = tid + it * 256;         // float4 index 0..2047
        int r   = idx >> 5;               // row 0..63
        int c4  = idx & 31;               // float4 col 0..31
        size_t b = (size_t)(b0 + r);
        v4f c  = *(const v4f*)&sC[r * CSTR + c4 * 4];
        v4f vj = *(const v4f*)(emb + (b * F_FEAT + fj) * D_DIM + c4 * 4);
        v4f o  = c * vj;
        v4f* dst = (v4f*)(out + (b * N_PAIRS + p) * (size_t)D_DIM + c4 * 4);
        __builtin_nontemporal_store(o, dst);   // NT: don't evict L2-resident W/emb
    }
}

extern "C" void kernel_launch(void* const* d_in, const int* in_sizes, int n_in,
                              void* d_out, int out_size, void* d_ws, size_t ws_size,
                              hipStream_t stream) {
    const float* emb = (const float*)d_in[0];   // (B, 30, 128) fp32
    const float* W   = (const float*)d_in[1];   // (435, 128, 128) fp32
    float* out = (float*)d_out;                 // (B, 435, 128) fp32

    const int B = in_sizes[0] / (F_FEAT * D_DIM);   // 4096
    dim3 grid(B / M_TILE, N_PAIRS);
    bilinear_wmma_kernel<<<grid, 256, 0, stream>>>(emb, W, out);
}